// VGRNN_51805895524407
// MI455X (gfx1250) — compile-verified
//
#include <hip/hip_runtime.h>
#include <math.h>

// ---------------------------------------------------------------------------
// VGRNN forward for MI455X (gfx1250). fp32 end-to-end; all dense GEMMs use
// V_WMMA_F32_16X16X4_F32 (wave32, 16x16 C tile per wave). Memory-bound parts
// (adj read + dec write, ~800MB @ 23.3TB/s) are single-pass streaming kernels.
// ---------------------------------------------------------------------------

#define LRELU_ALPHA 0.2f
#define T_STEPS 6
#define NN      4096
#define EE      65536
#define XDIM    128
#define HDIM    128
#define ZDIM    64
#define NPART   1024   // partial blocks for deterministic reductions

typedef float v2f __attribute__((ext_vector_type(2)));
typedef float v8f __attribute__((ext_vector_type(8)));

enum { ACT_NONE = 0, ACT_RELU, ACT_ELU, ACT_SOFTPLUS, ACT_SIGMOID };

__device__ __forceinline__ float actf(float x, int act) {
  switch (act) {
    case ACT_RELU:     return x > 0.f ? x : 0.f;
    case ACT_ELU:      return x > 0.f ? x : expm1f(x);
    case ACT_SOFTPLUS: return x > 20.f ? x : log1pf(expf(x));
    case ACT_SIGMOID:  return 1.f / (1.f + expf(-x));
    default:           return x;
  }
}

// ---------------------------------------------------------------------------
// GEMM: C[M,N] = act(A[M,K] @ W[K,N] + bias).  One wave per 16x16 tile,
// K stepped by 4 with v_wmma_f32_16x16x4_f32. Grid exact -> EXEC all ones.
// A 16x4 layout: lane=(k>=2)*16+m, vgpr=k&1.  B 4x16: lane=(k>=2)*16+n.
// C: vgpr r -> M = r + 8*(lane>>4), N = lane&15.
// ---------------------------------------------------------------------------
__global__ void gemm_wmma_kernel(const float* __restrict__ A,
                                 const float* __restrict__ W,
                                 const float* __restrict__ bias,
                                 float* __restrict__ C,
                                 int M, int K, int N, int act) {
  const int wave = threadIdx.x >> 5;
  const int lane = threadIdx.x & 31;
  const int row0 = (blockIdx.x * 4 + wave) * 16;
  const int col0 = blockIdx.y * 16;
  const int lm = lane & 15;   // m for A loads, n for B loads / C stores
  const int kh = lane >> 4;   // which K-pair / which C row-half

  v8f acc = {};
  const float* arow = A + (size_t)(row0 + lm) * K + 2 * kh;
  const float* bcol = W + (size_t)(2 * kh) * N + col0 + lm;
  for (int kb = 0; kb < K; kb += 4) {
    v2f a, b;
    a.x = arow[kb];
    a.y = arow[kb + 1];
    b.x = bcol[(size_t)kb * N];
    b.y = bcol[(size_t)(kb + 1) * N];
    acc = __builtin_amdgcn_wmma_f32_16x16x4_f32(false, a, false, b,
                                                (short)0, acc, false, false);
  }
  const float bn = bias ? bias[col0 + lm] : 0.f;
#pragma unroll
  for (int r = 0; r < 8; ++r) {
    const int m = r + 8 * kh;
    C[(size_t)(row0 + m) * N + col0 + lm] = actf(acc[r] + bn, act);
  }
}

// per-node attention scores: s1[i] = h[i] . a[0:F], s2[i] = h[i] . a[F:2F]
__global__ void node_scores_kernel(const float* __restrict__ h,
                                   const float* __restrict__ a,
                                   int F, float* __restrict__ s1,
                                   float* __restrict__ s2) {
  const int i = blockIdx.x * blockDim.x + threadIdx.x;
  if (i >= NN) return;
  float x1 = 0.f, x2 = 0.f;
  const float* hr = h + (size_t)i * F;
  for (int f = 0; f < F; ++f) {
    x1 += hr[f] * a[f];
    x2 += hr[f] * a[F + f];
  }
  s1[i] = x1;
  s2[i] = x2;
}

// one warp per edge (E real edges then N self-loops):
// w = exp(-leaky_relu(s1[src]+s2[dst])); rowsum[src]+=w; hp[src,:]+=w*h[dst,:]
__global__ void gat_edges_kernel(const int* __restrict__ esrc,
                                 const int* __restrict__ edst,
                                 const float* __restrict__ s1,
                                 const float* __restrict__ s2,
                                 const float* __restrict__ h, int F,
                                 float* __restrict__ rowsum,
                                 float* __restrict__ hp) {
  const int warp = (blockIdx.x * blockDim.x + threadIdx.x) >> 5;
  const int lane = threadIdx.x & 31;
  const int total = EE + NN;
  if (warp >= total) return;
  int src, dst;
  if (warp < EE) { src = esrc[warp]; dst = edst[warp]; }
  else           { src = dst = warp - EE; }
  const float sc = s1[src] + s2[dst];
  const float lr = sc > 0.f ? sc : LRELU_ALPHA * sc;
  const float w  = expf(-lr);
  if (lane == 0) atomicAdd(&rowsum[src], w);
  const float* hd = h + (size_t)dst * F;
  float* hs = hp + (size_t)src * F;
  for (int f = lane; f < F; f += 32) atomicAdd(&hs[f], w * hd[f]);
}

// out = act(hp / rowsum)  (optionally accumulated into out)
__global__ void gat_finalize_kernel(const float* __restrict__ hp,
                                    const float* __restrict__ rowsum,
                                    float* __restrict__ out, int F, int act,
                                    int accumulate) {
  const int idx = blockIdx.x * blockDim.x + threadIdx.x;
  if (idx >= NN * F) return;
  const float v = actf(hp[idx] / rowsum[idx / F], act);
  out[idx] = accumulate ? out[idx] + v : v;
}

__global__ void concat_rows_kernel(const float* __restrict__ A, int Fa,
                                   const float* __restrict__ B, int Fb,
                                   float* __restrict__ out) {
  const int F = Fa + Fb;
  const int idx = blockIdx.x * blockDim.x + threadIdx.x;
  if (idx >= NN * F) return;
  const int i = idx / F, f = idx - i * F;
  out[idx] = (f < Fa) ? A[(size_t)i * Fa + f] : B[(size_t)i * Fb + (f - Fa)];
}

// z = mu + eps * sig
__global__ void z_combine_kernel(const float* __restrict__ mu,
                                 const float* __restrict__ eps,
                                 const float* __restrict__ sig,
                                 float* __restrict__ z, int n) {
  const int i = blockIdx.x * blockDim.x + threadIdx.x;
  if (i < n) z[i] = mu[i] + eps[i] * sig[i];
}

// d1[i] = z[i].dec_a[0:Z], d2[i] = z[i].dec_a[Z:2Z]
__global__ void dec_scores_kernel(const float* __restrict__ z,
                                  const float* __restrict__ dec_a,
                                  float* __restrict__ d1,
                                  float* __restrict__ d2) {
  const int i = blockIdx.x * blockDim.x + threadIdx.x;
  if (i >= NN) return;
  float a1 = 0.f, a2 = 0.f;
  const float* zr = z + (size_t)i * ZDIM;
  for (int k = 0; k < ZDIM; ++k) {
    a1 += zr[k] * dec_a[k];
    a2 += zr[k] * dec_a[ZDIM + k];
  }
  d1[i] = a1;
  d2[i] = a2;
}

__device__ __forceinline__ float block_reduce_256(float a, float* sm) {
  sm[threadIdx.x] = a;
  __syncthreads();
  for (int s = 128; s > 0; s >>= 1) {
    if ((int)threadIdx.x < s) sm[threadIdx.x] += sm[threadIdx.x + s];
    __syncthreads();
  }
  return sm[0];
}

// deterministic partial sum (fixed strides); grid = NPART x 256
__global__ void sum_partial_kernel(const float* __restrict__ in, int n,
                                   float* __restrict__ partials) {
  __shared__ float sm[256];
  float a = 0.f;
  for (int i = blockIdx.x * 256 + threadIdx.x; i < n; i += NPART * 256)
    a += in[i];
  const float t = block_reduce_256(a, sm);
  if (threadIdx.x == 0) partials[blockIdx.x] = t;
}

__global__ void store_scalar_kernel(const float* __restrict__ partials,
                                    float* __restrict__ out) {
  __shared__ float sm[256];
  float a = 0.f;
  for (int i = threadIdx.x; i < NPART; i += 256) a += partials[i];
  const float t = block_reduce_256(a, sm);
  if (threadIdx.x == 0) out[0] = t;
}

// fused decoder + BCE: dec[i,j] = sigmoid(d1[i]+d2[j]); partial BCE sums
__global__ void dec_nll_partial_kernel(const float* __restrict__ d1,
                                       const float* __restrict__ d2,
                                       const float* __restrict__ adj,
                                       const float* __restrict__ adjsum,
                                       float* __restrict__ dec,
                                       float* __restrict__ partials) {
  __shared__ float sm[256];
  const float tot = (float)NN * (float)NN;
  const float s = adjsum[0];
  const float posw = (tot - s) / s;
  float acc = 0.f;
  const int total = NN * NN;
  for (int idx = blockIdx.x * 256 + threadIdx.x; idx < total;
       idx += NPART * 256) {
    const int i = idx >> 12;           // / 4096
    const int j = idx & (NN - 1);
    const float p = 1.f / (1.f + expf(-(d1[i] + d2[j])));
    dec[idx] = p;
    const float t = adj[idx];
    acc += -(posw * t * logf(p + 1e-7f) + (1.f - t) * logf(1.f - p + 1e-7f));
  }
  const float t = block_reduce_256(acc, sm);
  if (threadIdx.x == 0) partials[blockIdx.x] = t;
}

__global__ void nll_final_kernel(const float* __restrict__ partials,
                                 const float* __restrict__ adjsum,
                                 float* __restrict__ nll_out) {
  __shared__ float sm[256];
  float a = 0.f;
  for (int i = threadIdx.x; i < NPART; i += 256) a += partials[i];
  const float t = block_reduce_256(a, sm);
  if (threadIdx.x == 0) {
    const float tot = (float)NN * (float)NN;
    const float s = adjsum[0];
    const float norm = tot / ((tot - s) * 2.f);
    nll_out[0] += norm * (t / tot);
  }
}

__global__ void kld_partial_kernel(const float* __restrict__ mu,
                                   const float* __restrict__ sig,
                                   const float* __restrict__ pmu,
                                   const float* __restrict__ psig, int n,
                                   float* __restrict__ partials) {
  __shared__ float sm[256];
  const float eps = 1e-8f;
  float a = 0.f;
  for (int i = blockIdx.x * 256 + threadIdx.x; i < n; i += NPART * 256) {
    const float s1e = sig[i] + eps;
    const float s2e = psig[i] + eps;
    const float dm = mu[i] - pmu[i];
    a += 2.f * logf(s2e) - 2.f * logf(s1e) +
         (s1e * s1e + dm * dm) / (s2e * s2e) - 1.f;
  }
  const float t = block_reduce_256(a, sm);
  if (threadIdx.x == 0) partials[blockIdx.x] = t;
}

__global__ void kld_final_kernel(const float* __restrict__ partials,
                                 float* __restrict__ kld_out) {
  __shared__ float sm[256];
  float a = 0.f;
  for (int i = threadIdx.x; i < NPART; i += 256) a += partials[i];
  const float t = block_reduce_256(a, sm);
  if (threadIdx.x == 0)
    kld_out[0] += 0.5f * t / ((float)NN * (float)NN);
}

// gates: i=g0, f=g1, o=g2, c_tilde=g3
__global__ void lstm_combine_kernel(const float* __restrict__ g0,
                                    const float* __restrict__ g1,
                                    const float* __restrict__ g2,
                                    const float* __restrict__ g3,
                                    const float* __restrict__ c_prev,
                                    float* __restrict__ c_new,
                                    float* __restrict__ h_new, int n) {
  const int i = blockIdx.x * blockDim.x + threadIdx.x;
  if (i >= n) return;
  const float ig = 1.f / (1.f + expf(-g0[i]));
  const float fg = 1.f / (1.f + expf(-g1[i]));
  const float og = 1.f / (1.f + expf(-g2[i]));
  const float ct = tanhf(g3[i]);
  const float c = fg * c_prev[i] + ig * ct;
  c_new[i] = c;
  h_new[i] = og * tanhf(c);
}

// ---------------------------------------------------------------------------
// Host orchestration
// ---------------------------------------------------------------------------
namespace {

struct GatScratch {
  float *h, *s1, *s2, *rowsum, *hp;
  const int *esrc, *edst;
};

inline void run_gemm(hipStream_t st, const float* A, const float* W,
                     const float* bias, float* C, int K, int N, int act) {
  dim3 grid(NN / 64, N / 16);
  gemm_wmma_kernel<<<grid, 128, 0, st>>>(A, W, bias, C, NN, K, N, act);
}

inline void run_gat(hipStream_t st, const GatScratch& g, const float* x,
                    int Fin, int Fout, const float* W, const float* avec,
                    int act, float* out, int accumulate) {
  run_gemm(st, x, W, nullptr, g.h, Fin, Fout, ACT_NONE);
  node_scores_kernel<<<NN / 256, 256, 0, st>>>(g.h, avec, Fout, g.s1, g.s2);
  hipMemsetAsync(g.rowsum, 0, NN * sizeof(float), st);
  hipMemsetAsync(g.hp, 0, (size_t)NN * Fout * sizeof(float), st);
  const int warps = EE + NN;                 // 8 warps / 256-thread block
  gat_edges_kernel<<<(warps + 7) / 8, 256, 0, st>>>(g.esrc, g.edst, g.s1,
                                                    g.s2, g.h, Fout,
                                                    g.rowsum, g.hp);
  gat_finalize_kernel<<<(NN * Fout + 255) / 256, 256, 0, st>>>(
      g.hp, g.rowsum, out, Fout, act, accumulate);
}

}  // namespace

extern "C" void kernel_launch(void* const* d_in, const int* in_sizes, int n_in,
                              void* d_out, int out_size, void* d_ws,
                              size_t ws_size, hipStream_t stream) {
  (void)in_sizes; (void)n_in; (void)out_size; (void)ws_size;
  // ---- inputs (setup_inputs order) ----
  const float* x        = (const float*)d_in[0];
  const int*   edges    = (const int*)d_in[1];
  const float* adj      = (const float*)d_in[2];
  const float* eps_n    = (const float*)d_in[3];
  const float* phi_x_W  = (const float*)d_in[4];
  const float* phi_x_b  = (const float*)d_in[5];
  const float* phi_z_W  = (const float*)d_in[6];
  const float* phi_z_b  = (const float*)d_in[7];
  const float* enc_W    = (const float*)d_in[8];
  const float* enc_a    = (const float*)d_in[9];
  const float* enc_mu_W = (const float*)d_in[10];
  const float* enc_mu_a = (const float*)d_in[11];
  const float* enc_lv_W = (const float*)d_in[12];
  const float* enc_lv_a = (const float*)d_in[13];
  const float* prior_W  = (const float*)d_in[14];
  const float* prior_b  = (const float*)d_in[15];
  const float* prior_mu_W = (const float*)d_in[16];
  const float* prior_mu_b = (const float*)d_in[17];
  const float* prior_lv_W = (const float*)d_in[18];
  const float* prior_lv_b = (const float*)d_in[19];
  const float* dec_a    = (const float*)d_in[20];
  const float* lstm_Wx0 = (const float*)d_in[21];  // [4,256,128]
  const float* lstm_ax0 = (const float*)d_in[22];  // [4,256]
  const float* lstm_Wx1 = (const float*)d_in[23];  // [4,128,128]
  const float* lstm_ax1 = (const float*)d_in[24];  // [4,256]
  const float* lstm_Wh  = (const float*)d_in[25];  // [2,4,128,128]
  const float* lstm_ah  = (const float*)d_in[26];  // [2,4,256]

  // ---- outputs: [kld, nll, dec_all(T*N*N), mu_all(T*N*Z)] ----
  float* out_f = (float*)d_out;
  float* dec_base = out_f + 2;
  float* mu_base  = dec_base + (size_t)T_STEPS * NN * NN;
  hipMemsetAsync(out_f, 0, 2 * sizeof(float), stream);

  // ---- workspace carve-up ----
  float* w = (float*)d_ws;
  size_t off = 0;
  auto alloc = [&](size_t n) { float* p = w + off; off += n; return p; };
  const size_t NH = (size_t)NN * HDIM, NZ = (size_t)NN * ZDIM;
  float* gat_h  = alloc(NH);
  float* s1     = alloc(NN);
  float* s2     = alloc(NN);
  float* rowsum = alloc(NN);
  float* hp     = alloc(NH);
  float* phi_x  = alloc(NH);
  float* cat    = alloc(2 * NH);
  float* enc_t  = alloc(NH);
  float* sig    = alloc(NZ);
  float* prior  = alloc(NH);
  float* pmu    = alloc(NZ);
  float* psig   = alloc(NZ);
  float* zb     = alloc(NZ);
  float* phi_z  = alloc(NH);
  float* d1     = alloc(NN);
  float* d2     = alloc(NN);
  float* g[4];
  for (int k = 0; k < 4; ++k) g[k] = alloc(NH);
  float* hA[2] = {alloc(NH), alloc(NH)};
  float* hB[2] = {alloc(NH), alloc(NH)};
  float* cA[2] = {alloc(NH), alloc(NH)};
  float* cB[2] = {alloc(NH), alloc(NH)};
  float* partials = alloc(NPART);
  float* adjsum   = alloc(4);

  // zero initial LSTM state
  for (int l = 0; l < 2; ++l) {
    hipMemsetAsync(hA[l], 0, NH * sizeof(float), stream);
    hipMemsetAsync(cA[l], 0, NH * sizeof(float), stream);
  }
  float *h_prev[2] = {hA[0], hA[1]}, *h_next[2] = {hB[0], hB[1]};
  float *c_prev[2] = {cA[0], cA[1]}, *c_next[2] = {cB[0], cB[1]};

  GatScratch gs{gat_h, s1, s2, rowsum, hp, nullptr, nullptr};

  for (int t = 0; t < T_STEPS; ++t) {
    const float* x_t   = x + (size_t)t * NN * XDIM;
    const float* adj_t = adj + (size_t)t * NN * NN;
    const float* eps_t = eps_n + (size_t)t * NN * ZDIM;
    gs.esrc = edges + (size_t)t * 2 * EE;
    gs.edst = gs.esrc + EE;
    float* mu_t  = mu_base + (size_t)t * NN * ZDIM;
    float* dec_t = dec_base + (size_t)t * NN * NN;

    // phi_x = relu(x @ W + b)
    run_gemm(stream, x_t, phi_x_W, phi_x_b, phi_x, XDIM, HDIM, ACT_RELU);
    // enc_t = GAT(concat(phi_x, h[-1]), elu)
    concat_rows_kernel<<<(int)(2 * NH + 255) / 256, 256, 0, stream>>>(
        phi_x, HDIM, h_prev[1], HDIM, cat);
    run_gat(stream, gs, cat, 2 * HDIM, HDIM, enc_W, enc_a, ACT_ELU, enc_t, 0);
    // mu / sig
    run_gat(stream, gs, enc_t, HDIM, ZDIM, enc_mu_W, enc_mu_a, ACT_NONE, mu_t, 0);
    run_gat(stream, gs, enc_t, HDIM, ZDIM, enc_lv_W, enc_lv_a, ACT_SOFTPLUS, sig, 0);
    // prior path
    run_gemm(stream, h_prev[1], prior_W, prior_b, prior, HDIM, HDIM, ACT_ELU);
    run_gemm(stream, prior, prior_mu_W, prior_mu_b, pmu, HDIM, ZDIM, ACT_NONE);
    run_gemm(stream, prior, prior_lv_W, prior_lv_b, psig, HDIM, ZDIM, ACT_SOFTPLUS);
    // z, phi_z
    z_combine_kernel<<<(int)(NZ + 255) / 256, 256, 0, stream>>>(mu_t, eps_t,
                                                                sig, zb, (int)NZ);
    run_gemm(stream, zb, phi_z_W, phi_z_b, phi_z, ZDIM, HDIM, ACT_RELU);
    // decoder + NLL (needs sum(adj_t) first)
    dec_scores_kernel<<<NN / 256, 256, 0, stream>>>(zb, dec_a, d1, d2);
    sum_partial_kernel<<<NPART, 256, 0, stream>>>(adj_t, NN * NN, partials);
    store_scalar_kernel<<<1, 256, 0, stream>>>(partials, adjsum);
    dec_nll_partial_kernel<<<NPART, 256, 0, stream>>>(d1, d2, adj_t, adjsum,
                                                      dec_t, partials);
    nll_final_kernel<<<1, 256, 0, stream>>>(partials, adjsum, out_f + 1);
    // KLD
    kld_partial_kernel<<<NPART, 256, 0, stream>>>(mu_t, sig, pmu, psig,
                                                  (int)NZ, partials);
    kld_final_kernel<<<1, 256, 0, stream>>>(partials, out_f);

    // LSTM: input layer0 = concat(phi_x, phi_z)
    concat_rows_kernel<<<(int)(2 * NH + 255) / 256, 256, 0, stream>>>(
        phi_x, HDIM, phi_z, HDIM, cat);
    for (int l = 0; l < 2; ++l) {
      const float* inp = (l == 0) ? cat : h_next[0];
      const int Fin    = (l == 0) ? 2 * HDIM : HDIM;
      const float* Wx  = (l == 0) ? lstm_Wx0 : lstm_Wx1;
      const float* ax  = (l == 0) ? lstm_ax0 : lstm_ax1;
      for (int k = 0; k < 4; ++k) {
        run_gat(stream, gs, inp, Fin, HDIM, Wx + (size_t)k * Fin * HDIM,
                ax + (size_t)k * 2 * HDIM, ACT_NONE, g[k], 0);
        run_gat(stream, gs, h_prev[l], HDIM, HDIM,
                lstm_Wh + (size_t)(l * 4 + k) * HDIM * HDIM,
                lstm_ah + (size_t)(l * 4 + k) * 2 * HDIM, ACT_NONE, g[k], 1);
      }
      lstm_combine_kernel<<<(int)(NH + 255) / 256, 256, 0, stream>>>(
          g[0], g[1], g[2], g[3], c_prev[l], c_next[l], h_next[l], (int)NH);
    }
    for (int l = 0; l < 2; ++l) {
      float* tmp;
      tmp = h_prev[l]; h_prev[l] = h_next[l]; h_next[l] = tmp;
      tmp = c_prev[l]; c_prev[l] = c_next[l]; c_next[l] = tmp;
    }
  }
}